// SAGE_6571299963288
// MI455X (gfx1250) — compile-verified
//
#include <hip/hip_runtime.h>

#define DIMF 64

typedef __attribute__((ext_vector_type(2))) float v2f;
typedef __attribute__((ext_vector_type(8))) float v8f;

// ---------------- zeroing kernels ----------------
__global__ void zero4_kernel(float4* __restrict__ p, long n4) {
    long i = (long)blockIdx.x * blockDim.x + threadIdx.x;
    long stride = (long)gridDim.x * blockDim.x;
    float4 z; z.x = 0.f; z.y = 0.f; z.z = 0.f; z.w = 0.f;
    for (; i < n4; i += stride) p[i] = z;
}

__global__ void zero1_kernel(float* __restrict__ p, long n) {
    long i = (long)blockIdx.x * blockDim.x + threadIdx.x;
    long stride = (long)gridDim.x * blockDim.x;
    for (; i < n; i += stride) p[i] = 0.f;
}

// ---------------- degree ----------------
__global__ void degree_kernel(const int* __restrict__ dst, float* __restrict__ deg, int E) {
    int e = blockIdx.x * blockDim.x + threadIdx.x;
    if (e < E) {
        __hip_atomic_fetch_add(&deg[dst[e]], 1.0f,
                               __ATOMIC_RELAXED, __HIP_MEMORY_SCOPE_AGENT);
    }
}

__global__ void invdeg_kernel(float* __restrict__ deg, int N) {
    int i = blockIdx.x * blockDim.x + threadIdx.x;
    if (i < N) deg[i] = 1.0f / fmaxf(deg[i], 1.0f);
}

// ---------------- edge aggregation: summed[dst] += feat[src] ----------------
// 16 threads per edge, float4 per thread (64 floats per row).
__global__ void agg_kernel(const float4* __restrict__ feat4,
                           const int* __restrict__ src,
                           const int* __restrict__ dst,
                           float* __restrict__ summed, int E) {
    int tid = blockIdx.x * blockDim.x + threadIdx.x;
    int e = tid >> 4;
    int part = tid & 15;
    if (e >= E) return;
    int s = src[e];
    int d = dst[e];
    float4 v = feat4[(long)s * 16 + part];
    float* o = summed + (long)d * DIMF + part * 4;
    __hip_atomic_fetch_add(o + 0, v.x, __ATOMIC_RELAXED, __HIP_MEMORY_SCOPE_AGENT);
    __hip_atomic_fetch_add(o + 1, v.y, __ATOMIC_RELAXED, __HIP_MEMORY_SCOPE_AGENT);
    __hip_atomic_fetch_add(o + 2, v.z, __ATOMIC_RELAXED, __HIP_MEMORY_SCOPE_AGENT);
    __hip_atomic_fetch_add(o + 3, v.w, __ATOMIC_RELAXED, __HIP_MEMORY_SCOPE_AGENT);
}

// ---------------- fused SAGE GEMM ----------------
// out[n,:] = act( x[n,:] @ Wself + (summed[n,:]*invdeg[n]) @ Wneigh + b )
// Block: 256 threads = 8 waves = 2 Mtiles(16 rows) x 4 Ntiles(16 cols), K=128
// via V_WMMA_F32_16X16X4_F32 chains.
#define WST 80   // weight LDS row stride (floats): bank groups disjoint
#define AST 132  // A-tile LDS row stride (floats): conflict-free b64 reads

template<bool RELU>
__global__ __launch_bounds__(256)
void sage_gemm_kernel(const float* __restrict__ x,
                      const float* __restrict__ summed,
                      const float* __restrict__ invdeg,
                      const float* __restrict__ Wself,
                      const float* __restrict__ Wneigh,
                      const float* __restrict__ bias,
                      float* __restrict__ out, int N) {
    __shared__ float sW[2 * 64 * WST];  // [0]=Wself, [1]=Wneigh  (40960 B)
    __shared__ float sA[32 * AST];      // 32 rows x (64 self | 64 mean) (16896 B)

    const int t    = threadIdx.x;
    const int lane = t & 31;
    const int wave = t >> 5;
    const int mt   = wave >> 2;   // 0..1  M-tile of this wave
    const int nt   = wave & 3;    // 0..3  N-tile of this wave
    const int g    = lane >> 4;   // half-wave group (0/1)
    const int ln   = lane & 15;

    // Stage both weight matrices once per block (row-major K x 64).
    for (int i = t; i < 64 * 16; i += 256) {
        int row = i >> 4, q = i & 15;
        float4 vs = ((const float4*)Wself)[i];
        float4 vn = ((const float4*)Wneigh)[i];
        float* ps = &sW[row * WST + q * 4];
        ps[0] = vs.x; ps[1] = vs.y; ps[2] = vs.z; ps[3] = vs.w;
        float* pn = &sW[64 * WST + row * WST + q * 4];
        pn[0] = vn.x; pn[1] = vn.y; pn[2] = vn.z; pn[3] = vn.w;
    }
    const float bval = bias[nt * 16 + ln];

    const int ntiles = (N + 31) >> 5;
    for (int tile = blockIdx.x; tile < ntiles; tile += gridDim.x) {
        const int base = tile * 32;
        __syncthreads();  // previous iteration's sA reads done
        // Stage A tile: 32 rows x 128 cols (self | mean), float4 granularity.
        for (int i = t; i < 1024; i += 256) {
            int row = i >> 5;          // 0..31
            int q   = i & 31;          // float4 slot
            int node = base + row;
            float4 v; v.x = 0.f; v.y = 0.f; v.z = 0.f; v.w = 0.f;
            int col;
            if (q < 16) {
                col = q * 4;
                if (node < N) v = ((const float4*)x)[(long)node * 16 + q];
            } else {
                col = 64 + (q - 16) * 4;
                if (node < N) {
                    v = ((const float4*)summed)[(long)node * 16 + (q - 16)];
                    float id = invdeg[node];
                    v.x *= id; v.y *= id; v.z *= id; v.w *= id;
                }
            }
            float* p = &sA[row * AST + col];
            p[0] = v.x; p[1] = v.y; p[2] = v.z; p[3] = v.w;
        }
        __syncthreads();

        v8f acc = {bval, bval, bval, bval, bval, bval, bval, bval};
        const int arow = mt * 16 + ln;
        const float* __restrict__ aRow = &sA[arow * AST];
#pragma unroll
        for (int k = 0; k < 32; ++k) {
            // A fragment: lanes<16 hold K={4k,4k+1}; lanes>=16 hold K={4k+2,4k+3}
            int kk = 4 * k + 2 * g;
            v2f a;
            a.x = aRow[kk];
            a.y = aRow[kk + 1];
            // B fragment: same K rows of the matching weight matrix
            const float* W = (k < 16) ? sW : (sW + 64 * WST);
            int kb = (k & 15) * 4 + 2 * g;
            v2f b;
            b.x = W[kb * WST + nt * 16 + ln];
            b.y = W[(kb + 1) * WST + nt * 16 + ln];
            acc = __builtin_amdgcn_wmma_f32_16x16x4_f32(
                false, a, false, b, (short)0, acc, false, false);
        }

        // D layout: lanes<16 -> M=v, N=lane ; lanes>=16 -> M=v+8, N=lane-16
        const int col   = nt * 16 + ln;
        const int rbase = base + mt * 16 + g * 8;
#pragma unroll
        for (int v = 0; v < 8; ++v) {
            int r = rbase + v;
            if (r < N) {
                float val = acc[v];
                if (RELU) val = fmaxf(val, 0.0f);
                out[(long)r * DIMF + col] = val;
            }
        }
    }
}

// ---------------- launcher ----------------
extern "C" void kernel_launch(void* const* d_in, const int* in_sizes, int n_in,
                              void* d_out, int out_size, void* d_ws, size_t ws_size,
                              hipStream_t stream) {
    const float* x   = (const float*)d_in[0];
    const int*   src = (const int*)d_in[1];
    const int*   dst = (const int*)d_in[2];
    const float* Ws1 = (const float*)d_in[3];
    const float* Wn1 = (const float*)d_in[4];
    const float* b1  = (const float*)d_in[5];
    const float* Ws2 = (const float*)d_in[6];
    const float* Wn2 = (const float*)d_in[7];
    const float* b2  = (const float*)d_in[8];
    float* out = (float*)d_out;

    const int N = in_sizes[0] / DIMF;
    const int E = in_sizes[1];

    float* summed = (float*)d_ws;                 // N*64 f32
    float* h      = summed + (size_t)N * DIMF;    // N*64 f32
    float* deg    = h + (size_t)N * DIMF;         // N f32 (becomes invdeg)

    const long feat4 = (long)N * DIMF / 4;
    const int  aggBlocks  = (int)(((long)E * 16 + 255) / 256);
    const int  gemmBlocks = 1024;

    // degree -> invdeg (shared by both layers)
    zero1_kernel<<<256, 256, 0, stream>>>(deg, N);
    degree_kernel<<<(E + 255) / 256, 256, 0, stream>>>(dst, deg, E);
    invdeg_kernel<<<(N + 255) / 256, 256, 0, stream>>>(deg, N);

    // ---- layer 1: h = relu(x@Ws1 + b1 + mean(x)@Wn1)
    zero4_kernel<<<1024, 256, 0, stream>>>((float4*)summed, feat4);
    agg_kernel<<<aggBlocks, 256, 0, stream>>>((const float4*)x, src, dst, summed, E);
    sage_gemm_kernel<true><<<gemmBlocks, 256, 0, stream>>>(
        x, summed, deg, Ws1, Wn1, b1, h, N);

    // ---- layer 2: out = h@Ws2 + b2 + mean(h)@Wn2
    zero4_kernel<<<1024, 256, 0, stream>>>((float4*)summed, feat4);
    agg_kernel<<<aggBlocks, 256, 0, stream>>>((const float4*)h, src, dst, summed, E);
    sage_gemm_kernel<false><<<gemmBlocks, 256, 0, stream>>>(
        h, summed, deg, Ws2, Wn2, b2, out, N);
}